// TopkRouter_63161789054986
// MI455X (gfx1250) — compile-verified
//
#include <hip/hip_runtime.h>
#include <stdint.h>

#define D_MODEL 4096
#define N_EXPERTS 64
#define N_TOKENS 16384

#define KC 64                    // K chunk staged in LDS
#define NCHUNK (D_MODEL / KC)    // 64
#define LDSS 68                  // padded LDS row stride (floats): 272B = 16B-aligned, conflict-free
#define WAVES 4                  // waves per block
#define TPB (WAVES * 32)

typedef __attribute__((ext_vector_type(2))) float v2f;
typedef __attribute__((ext_vector_type(8))) float v8f;

// ---- CDNA5 async memory->LDS copy (ASYNCcnt-tracked, no VGPR round trip) ----
__device__ __forceinline__ void async_load_b128(uint32_t lds_off, uint64_t gaddr) {
    // GLOBAL_LOAD_ASYNC_TO_LDS_B128: VDST = LDS byte offset, VADDR = 64-bit global addr (GV mode)
    asm volatile("global_load_async_to_lds_b128 %0, %1, off"
                 :: "v"(lds_off), "v"(gaddr)
                 : "memory");
}

template <int N>
__device__ __forceinline__ void wait_asynccnt() {
#if __has_builtin(__builtin_amdgcn_s_wait_asynccnt)
    __builtin_amdgcn_s_wait_asynccnt(N);
#else
    asm volatile("s_wait_asynccnt %0" :: "n"(N) : "memory");
#endif
}

__device__ __forceinline__ uint32_t lds_addr(const void* p) {
    // generic pointer to LDS: low 32 bits hold the LDS byte offset
    return (uint32_t)(uint64_t)p;
}

__device__ __forceinline__ bool better(float va, int ia, float vb, int ib) {
    // jax.lax.top_k ordering: larger value first; ties -> lower index first
    return (va > vb) || (va == vb && ia < ib);
}

__global__ __launch_bounds__(TPB)
void TopkRouter_63161789054986_kernel(const float* __restrict__ x,
                                      const float* __restrict__ W,
                                      const float* __restrict__ bias,
                                      float* __restrict__ out,
                                      int* __restrict__ idx_out) {
    __shared__ __align__(16) float sW[2][N_EXPERTS][LDSS];   // double-buffered 64x64 W chunk
    __shared__ __align__(16) float sX[2][WAVES][16][LDSS];   // double-buffered per-wave 16x64 x chunk

    const int tid   = threadIdx.x;
    const int wave  = tid >> 5;
    const int lane  = tid & 31;
    const int laneH = lane & 15;
    const int half  = lane >> 4;              // 0: K low half, 1: K high half
    const int m0    = blockIdx.x * (WAVES * 16) + wave * 16;

    v8f acc0 = {}, acc1 = {}, acc2 = {}, acc3 = {};

    // Issue one K-chunk of async copies: 16 B128 ops per wave (8 for W, 8 for x).
    auto issue_chunk = [&](int buf, int k0) {
#pragma unroll
        for (int it = 0; it < 8; ++it) {      // W[0:64][k0:k0+64], block-cooperative
            int idx = it * TPB + tid;         // 0..1023 float4 slots
            int e = idx >> 4, c = idx & 15;
            async_load_b128(lds_addr(&sW[buf][e][c * 4]),
                            (uint64_t)(W + (size_t)e * D_MODEL + k0 + c * 4));
        }
#pragma unroll
        for (int it = 0; it < 8; ++it) {      // x[m0:m0+16][k0:k0+64], per-wave
            int idx = it * 32 + lane;         // 0..255 float4 slots
            int row = idx >> 4, c = idx & 15;
            async_load_b128(lds_addr(&sX[buf][wave][row][c * 4]),
                            (uint64_t)(x + (size_t)(m0 + row) * D_MODEL + k0 + c * 4));
        }
    };

    issue_chunk(0, 0);                        // prime the pipeline

    for (int i = 0; i < NCHUNK; ++i) {
        const int buf = i & 1;
        if (i + 1 < NCHUNK) {
            issue_chunk(buf ^ 1, (i + 1) * KC);   // prefetch next chunk into other buffer
            wait_asynccnt<16>();                  // in-order completion: chunk i's 16 ops landed
        } else {
            wait_asynccnt<0>();
        }
        __syncthreads();                          // all waves' deposits visible

        // --- 16 K-steps of V_WMMA_F32_16X16X4_F32 across 4 expert tiles ---
#pragma unroll
        for (int kk = 0; kk < KC; kk += 4) {
            const int col = kk + 2 * half;        // lanes 0-15: K={kk,kk+1}; 16-31: K={kk+2,kk+3}
            v2f a  = *(const v2f*)&sX[buf][wave][laneH][col];
            v2f b0 = *(const v2f*)&sW[buf][ 0 + laneH][col];
            v2f b1 = *(const v2f*)&sW[buf][16 + laneH][col];
            v2f b2 = *(const v2f*)&sW[buf][32 + laneH][col];
            v2f b3 = *(const v2f*)&sW[buf][48 + laneH][col];
            acc0 = __builtin_amdgcn_wmma_f32_16x16x4_f32(false, a, false, b0, (short)0, acc0, false, false);
            acc1 = __builtin_amdgcn_wmma_f32_16x16x4_f32(false, a, false, b1, (short)0, acc1, false, false);
            acc2 = __builtin_amdgcn_wmma_f32_16x16x4_f32(false, a, false, b2, (short)0, acc2, false, false);
            acc3 = __builtin_amdgcn_wmma_f32_16x16x4_f32(false, a, false, b3, (short)0, acc3, false, false);
        }
        __syncthreads();                          // readers done: next iter may overwrite this buffer
    }

    // --- bias ---
    const float bv0 = bias[ 0 + laneH];
    const float bv1 = bias[16 + laneH];
    const float bv2 = bias[32 + laneH];
    const float bv3 = bias[48 + laneH];

    // --- per-token top-2 + sparse softmax ---
    // C/D layout: vgpr r of acc holds token m0 + r + 8*half, expert t*16 + laneH.
#pragma unroll
    for (int r = 0; r < 8; ++r) {
        const int token = m0 + r + 8 * half;
        const float c0 = acc0[r] + bv0;
        const float c1 = acc1[r] + bv1;
        const float c2 = acc2[r] + bv2;
        const float c3 = acc3[r] + bv3;

        float v1 = c0, v2 = -__builtin_inff();
        int   i1 = laneH, i2 = 0x7fffffff;
        auto ins = [&](float cv, int ci) {
            if (better(cv, ci, v1, i1)) { v2 = v1; i2 = i1; v1 = cv; i1 = ci; }
            else if (better(cv, ci, v2, i2)) { v2 = cv; i2 = ci; }
        };
        ins(c1, 16 + laneH);
        ins(c2, 32 + laneH);
        ins(c3, 48 + laneH);

        // merge sorted pairs across the 16-lane half-group (xor masks stay in-group)
#pragma unroll
        for (int m = 1; m <= 8; m <<= 1) {
            float ov1 = __shfl_xor(v1, m, 32);
            int   oi1 = __shfl_xor(i1, m, 32);
            float ov2 = __shfl_xor(v2, m, 32);
            int   oi2 = __shfl_xor(i2, m, 32);
            if (better(ov1, oi1, v1, i1)) {
                float nv2; int ni2;
                if (better(v1, i1, ov2, oi2)) { nv2 = v1; ni2 = i1; }
                else                          { nv2 = ov2; ni2 = oi2; }
                v1 = ov1; i1 = oi1; v2 = nv2; i2 = ni2;
            } else if (better(ov1, oi1, v2, i2)) {
                v2 = ov1; i2 = oi1;
            }
        }

        // sparse softmax over {v1, v2}; all other experts get 0
        const float e2 = __expf(v2 - v1);       // v2 <= v1, arg <= 0: safe
        const float p1 = 1.0f / (1.0f + e2);
        const float p2 = 1.0f - p1;

        const int e0 = laneH, eA = 16 + laneH, eB = 32 + laneH, eC = 48 + laneH;
        float* orow = out + (size_t)token * N_EXPERTS;
        orow[e0] = (e0 == i1) ? p1 : ((e0 == i2) ? p2 : 0.0f);
        orow[eA] = (eA == i1) ? p1 : ((eA == i2) ? p2 : 0.0f);
        orow[eB] = (eB == i1) ? p1 : ((eB == i2) ? p2 : 0.0f);
        orow[eC] = (eC == i1) ? p1 : ((eC == i2) ? p2 : 0.0f);
        if (laneH == 0) {
            idx_out[2 * token + 0] = i1;
            idx_out[2 * token + 1] = i2;
        }
    }
}

extern "C" void kernel_launch(void* const* d_in, const int* in_sizes, int n_in,
                              void* d_out, int out_size, void* d_ws, size_t ws_size,
                              hipStream_t stream) {
    (void)in_sizes; (void)n_in; (void)d_ws; (void)ws_size; (void)out_size;
    const float* x = (const float*)d_in[0];
    const float* W = (const float*)d_in[1];
    const float* b = (const float*)d_in[2];
    float* out = (float*)d_out;
    int* idx = (int*)((float*)d_out + (size_t)N_TOKENS * N_EXPERTS);

    dim3 grid(N_TOKENS / (WAVES * 16));   // 256 blocks x 128 threads (4 waves)
    TopkRouter_63161789054986_kernel<<<grid, TPB, 0, stream>>>(x, W, b, out, idx);
}